// GraphLearning_58935541236191
// MI455X (gfx1250) — compile-verified
//
#include <hip/hip_runtime.h>
#include <hip/hip_bf16.h>

typedef __attribute__((ext_vector_type(2))) float        v2f;
typedef __attribute__((ext_vector_type(8))) float        v8f;
typedef __attribute__((ext_vector_type(4))) unsigned int v4u;
typedef __attribute__((ext_vector_type(8))) int          v8i;
typedef __attribute__((ext_vector_type(4))) int          v4i;

// Problem sizes from the reference setup_inputs()
#define NROWS 8192
#define DDIM  256
#define DX    64
#define KSEL  10

// d_out layout (floats), concatenated in reference return order
#define OUT_X      0
#define OUT_EDGES  524288            // 2 * 81920
#define OUT_PHEN   688128
#define OUT_LOGP   2785280
#define OUT_ATT    2867200

#define ASTRIDE 258                  // padded LDS row stride (floats), bank-conflict free

// ---------------------------------------------------------------------------
// sorted top-K (smallest) insertion into a register-resident list
// ---------------------------------------------------------------------------
__device__ __forceinline__ void tk_insert(float v, int j, float (&vals)[KSEL], int (&idxs)[KSEL]) {
    if (v >= vals[KSEL - 1]) return;
    vals[KSEL - 1] = v;
    idxs[KSEL - 1] = j;
#pragma unroll
    for (int t = KSEL - 1; t > 0; --t) {
        float va = vals[t - 1], vb = vals[t];
        int   ia = idxs[t - 1], ib = idxs[t];
        bool  sw = vb < va;
        vals[t - 1] = sw ? vb : va;  vals[t] = sw ? va : vb;
        idxs[t - 1] = sw ? ib : ia;  idxs[t] = sw ? ia : ib;
    }
}

// ---------------------------------------------------------------------------
// Kernel 1: att = sigmoid(phen @ W + bias); p = att * phen
// One wave per 16x16 tile; 64x V_WMMA_F32_16X16X4_F32 (f32 precision).
// ---------------------------------------------------------------------------
__global__ __launch_bounds__(256) void gemm1_att_kernel(
    const float* __restrict__ phen, const float* __restrict__ W,
    const float* __restrict__ bias, float* __restrict__ att_out,
    float* __restrict__ p_out) {
    const int lane = threadIdx.x & 31;
    const int wave = threadIdx.x >> 5;
    const int tile = blockIdx.x * 8 + wave;        // 8192 tiles total
    const int tm   = tile >> 4;                     // row block 0..511
    const int tn   = tile & 15;                     // col block 0..15
    const int lo   = lane & 15;
    const int hi   = lane >> 4;
    const int koff = 2 * hi;                        // A/B K-half select
    const int gm0  = tm * 16;
    const int gn   = tn * 16 + lo;

    v8f c = {};
    const float* arow = phen + (size_t)(gm0 + lo) * DDIM;
#pragma unroll 4
    for (int k0 = 0; k0 < DDIM; k0 += 4) {
        v2f a = *(const v2f*)(arow + k0 + koff);                 // A[M=lo][k0+koff..+1]
        v2f b;
        b.x = W[(size_t)(k0 + koff)     * DDIM + gn];            // B[K][N=gn]
        b.y = W[(size_t)(k0 + koff + 1) * DDIM + gn];
        c = __builtin_amdgcn_wmma_f32_16x16x4_f32(false, a, false, b, (short)0, c, false, false);
    }

    const float bn = bias[gn];
#pragma unroll
    for (int r = 0; r < 8; ++r) {
        int   m  = gm0 + r + 8 * hi;                             // C: M = r + 8*hi, N = lo
        float z  = c[r] + bn;
        float a  = __builtin_amdgcn_rcpf(1.0f + __expf(-z));     // v_rcp_f32 sigmoid
        float ph = phen[(size_t)m * DDIM + gn];
        att_out[(size_t)m * DDIM + gn] = a;
        p_out[(size_t)m * DDIM + gn]   = a * ph;
    }
}

// ---------------------------------------------------------------------------
// Kernel 2: sq[m] = sum_d p[m][d]^2
// ---------------------------------------------------------------------------
__global__ __launch_bounds__(256) void rowsq_kernel(const float* __restrict__ p,
                                                    float* __restrict__ sq) {
    __shared__ float red[256];
    const int row = blockIdx.x;
    float v = p[(size_t)row * DDIM + threadIdx.x];
    red[threadIdx.x] = v * v;
    __syncthreads();
#pragma unroll
    for (int s = 128; s > 0; s >>= 1) {
        if (threadIdx.x < s) red[threadIdx.x] += red[threadIdx.x + s];
        __syncthreads();
    }
    if (threadIdx.x == 0) sq[row] = red[0];
}

// ---------------------------------------------------------------------------
// Kernel 3: fused  D = sq_i + sq_j - 2 p p^T  ->  lq  ->  per-row Gumbel top-10.
// A-strip (16 contiguous rows of p = one contiguous 16 KB block) is DMA'd to
// LDS by the Tensor Data Mover with pad_enable inserting 2 DWORDs every 256
// DWORDs -> the 258-float padded stride that kills A-read bank conflicts.
// Each wave handles two adjacent 16-col tiles per iteration (A-frag reuse,
// independent WMMA chains), transposes lq through a private LDS slab, and
// maintains per-lane register top-10 lists; final cross-wave merge in LDS.
// ---------------------------------------------------------------------------
__global__ __launch_bounds__(256) void dist_topk_kernel(
    const float* __restrict__ p, const float* __restrict__ sq,
    const float* __restrict__ q, const float* __restrict__ temp,
    float* __restrict__ logp_out, float* __restrict__ edges_out) {
    __shared__ __align__(16) float lA[16 * ASTRIDE + 8];  // TDM-padded A strip
    __shared__ float lsq[16];                 // row norms for this strip
    __shared__ float stage[8][16][33];        // per-wave 16x32 lq transpose slab
    __shared__ float cval[16][16][KSEL];      // [row][list][k] candidate values
    __shared__ int   cidx[16][16][KSEL];      // [row][list][k] candidate indices

    const int tid  = threadIdx.x;
    const int lane = tid & 31;
    const int wave = tid >> 5;
    const int lo   = lane & 15;
    const int hi   = lane >> 4;
    const int koff = 2 * hi;
    const int row0 = blockIdx.x * 16;

    // ---- TDM: one wave issues the 16KB contiguous A-strip DMA into LDS ----
    if (wave == 0) {
        unsigned long long ga = (unsigned long long)(p + (size_t)row0 * DDIM);
        unsigned int ldsa = (unsigned int)(unsigned long long)(const void*)&lA[0]; // low 32b = LDS offset

        v4u g0;
        g0[0] = 1u;                                   // count=1, user descriptor
        g0[1] = ldsa;                                 // lds_addr (bytes)
        g0[2] = (unsigned int)(ga & 0xffffffffull);   // global_addr[31:0]
        g0[3] = (unsigned int)((ga >> 32) & 0x1ffffffull) | 0x80000000u; // addr[56:32] | type=2

        v8i g1;
        g1[0] = (2 << 16)        // data_size = 4 bytes
              | (1 << 20)        // pad_enable
              | (7 << 22)        // pad_interval: every 256 DWORDs
              | (1 << 25);       // pad_amount: 2 DWORDs  -> 258-float row stride
        g1[1] = (int)(4096u << 16);  // tensor_dim0[15:0]  (dim0 @ bits 79:48)
        g1[2] = (int)(1u << 16);     // tensor_dim0[31:16]=0 | tensor_dim1[15:0]=1
        g1[3] = (int)(4096u << 16);  // tensor_dim1[31:16]=0 | tile_dim0=4096
        g1[4] = 1;                   // tile_dim1=1, tile_dim2=0
        g1[5] = 4096;                // tensor_dim0_stride[31:0]
        g1[6] = 0;                   // dim0_stride hi | dim1_stride lo
        g1[7] = 0;

        v4i g2;
        g2[0] = 1;                   // tensor_dim2 = 1
        g2[1] = 1;                   // tensor_dim3 = 1
        g2[2] = 4096;                // tensor_dim2_stride lo
        g2[3] = 0;                   // stride hi | tile_dim3 = 0

        v4i g3;
        g3[0] = 4096;                // tensor_dim3_stride lo
        g3[1] = (int)(1u << 16);     // tensor_dim4 = 1
        g3[2] = 0;                   // tile_dim4 = 0
        g3[3] = 0;

        v8i g4 = {0, 0, 0, 0, 0, 0, 0, 0};   // trailing group (6-arg toolchain form)

        __builtin_amdgcn_tensor_load_to_lds(g0, g1, g2, g3, g4, 0);
        __builtin_amdgcn_s_wait_tensorcnt(0);
    }
    if (tid < 16) lsq[tid] = sq[row0 + tid];
    __syncthreads();

    const float expT = __expf(fminf(fmaxf(temp[0], -5.0f), 5.0f));

    float vals[KSEL];
    int   idxs[KSEL];
#pragma unroll
    for (int t = 0; t < KSEL; ++t) { vals[t] = 3.4e38f; idxs[t] = 0; }
    const int selRow = lo;                    // local row this lane selects for

    for (int cp = wave; cp < NROWS / 32; cp += 8) {   // 32-column panel per iteration
        const int col0 = cp * 32;
        const float* bcol0 = p + (size_t)(col0 + lo) * DDIM;
        const float* bcol1 = bcol0 + (size_t)16 * DDIM;
        if (cp + 8 < NROWS / 32)              // warm caches for this wave's next panel
            __builtin_prefetch(p + (size_t)(col0 + 256 + lo) * DDIM, 0, 1);

        v8f c0 = {}, c1 = {};
#pragma unroll 4
        for (int k0 = 0; k0 < DDIM; k0 += 4) {
            v2f a  = *(const v2f*)(&lA[lo * ASTRIDE + k0 + koff]);  // A[M=lo][K] (shared)
            v2f b0 = *(const v2f*)(bcol0 + k0 + koff);              // B tile 0
            v2f b1 = *(const v2f*)(bcol1 + k0 + koff);              // B tile 1
            c0 = __builtin_amdgcn_wmma_f32_16x16x4_f32(false, a, false, b0, (short)0, c0, false, false);
            c1 = __builtin_amdgcn_wmma_f32_16x16x4_f32(false, a, false, b1, (short)0, c1, false, false);
        }

        // lq for both 16x16 tiles -> this wave's private 16x32 transpose slab
        const float sqc0 = sq[col0 + lo];
        const float sqc1 = sq[col0 + 16 + lo];
#pragma unroll
        for (int r = 0; r < 8; ++r) {
            int    m   = r + 8 * hi;                                // local row of element
            size_t qrb = (size_t)(row0 + m) * NROWS + col0;

            float D0 = fmaxf(lsq[m] + sqc0 - 2.0f * c0[r], 0.0f);
            float g0v = __logf(-__logf(q[qrb + lo] + 1e-8f));
            stage[wave][m][lo] = D0 * expT - g0v;

            float D1 = fmaxf(lsq[m] + sqc1 - 2.0f * c1[r], 0.0f);
            float g1v = __logf(-__logf(q[qrb + 16 + lo] + 1e-8f));
            stage[wave][m][16 + lo] = D1 * expT - g1v;
        }
        __builtin_amdgcn_wave_barrier();                            // order DS write -> DS read

        // lane (lo, hi) scans row `lo`, columns hi*16 .. hi*16+15 of the slab
#pragma unroll
        for (int cc = 0; cc < 16; ++cc) {
            int   cl = hi * 16 + cc;
            float v  = stage[wave][selRow][cl];
            tk_insert(v, col0 + cl, vals, idxs);
        }
        __builtin_amdgcn_wave_barrier();                            // reads before next writes
    }

    // dump per-lane lists: 16 lists per row (8 waves x 2 lane-halves)
    const int listId = wave * 2 + hi;
#pragma unroll
    for (int t = 0; t < KSEL; ++t) {
        cval[selRow][listId][t] = vals[t];
        cidx[selRow][listId][t] = idxs[t];
    }
    __syncthreads();

    // final merge: thread t (< 16) owns row t, merges 16 lists x 10 candidates
    if (tid < 16) {
        float fv[KSEL];
        int   fi[KSEL];
#pragma unroll
        for (int t = 0; t < KSEL; ++t) { fv[t] = 3.4e38f; fi[t] = 0; }
        for (int L = 0; L < 16; ++L)
#pragma unroll
            for (int t = 0; t < KSEL; ++t)
                tk_insert(cval[tid][L][t], cidx[tid][L][t], fv, fi);

        const int grow = row0 + tid;
#pragma unroll
        for (int t = 0; t < KSEL; ++t) {
            // top_k(-lq) returns descending -lq == ascending lq (our list order)
            logp_out[(size_t)grow * KSEL + t] = -fv[t];
            edges_out[(size_t)grow * KSEL + t]                  = (float)fi[t];  // indices row
            edges_out[(size_t)(NROWS * KSEL) + grow * KSEL + t] = (float)grow;   // source rows
        }
    }
}

// ---------------------------------------------------------------------------
// launch
// ---------------------------------------------------------------------------
extern "C" void kernel_launch(void* const* d_in, const int* in_sizes, int n_in,
                              void* d_out, int out_size, void* d_ws, size_t ws_size,
                              hipStream_t stream) {
    const float* x    = (const float*)d_in[0];
    // d_in[1] (A) is unused by the reference
    const float* phen = (const float*)d_in[2];
    const float* W    = (const float*)d_in[3];
    const float* bias = (const float*)d_in[4];
    const float* temp = (const float*)d_in[5];
    const float* q    = (const float*)d_in[6];
    // d_in[7] (k) == 10, compile-time KSEL

    float* out = (float*)d_out;

    // scratch: p [8192x256] then sq [8192]
    float* p_ws  = (float*)d_ws;
    float* sq_ws = p_ws + (size_t)NROWS * DDIM;

    // passthrough outputs (graph-capture-safe async d2d copies)
    (void)hipMemcpyAsync(out + OUT_X, x, (size_t)NROWS * DX * sizeof(float),
                         hipMemcpyDeviceToDevice, stream);
    (void)hipMemcpyAsync(out + OUT_PHEN, phen, (size_t)NROWS * DDIM * sizeof(float),
                         hipMemcpyDeviceToDevice, stream);

    // att + weighted phenotypes (WMMA f32)
    gemm1_att_kernel<<<(NROWS / 16) * (DDIM / 16) / 8, 256, 0, stream>>>(
        phen, W, bias, out + OUT_ATT, p_ws);

    // row squared norms
    rowsq_kernel<<<NROWS, 256, 0, stream>>>(p_ws, sq_ws);

    // fused distances + Gumbel top-k (WMMA f32 + TDM A-strip staging)
    dist_topk_kernel<<<NROWS / 16, 256, 0, stream>>>(
        p_ws, sq_ws, q, temp, out + OUT_LOGP, out + OUT_EDGES);
}